// SHEncoder_5669356832962
// MI455X (gfx1250) — compile-verified
//
#include <hip/hip_runtime.h>
#include <hip/hip_bf16.h>
#include <stdint.h>
#include <stddef.h>

// Real SH basis constants (degree 4 => l=0..3, 16 coefficients)
#define SH_C0    0.28209479177387814f
#define SH_C1    0.48860251190291987f
#define SH_C2_0  1.0925484305920792f
#define SH_C2_2a 0.94617469575755997f
#define SH_C2_2b 0.31539156525251999f
#define SH_C2_4  0.54627421529603959f
#define SH_C3_0  0.59004358992664352f
#define SH_C3_1  2.8906114426405538f
#define SH_C3_2  0.45704579946446572f
#define SH_C3_3  0.3731763325901154f
#define SH_C3_5  1.445305721320277f

// gfx1250 async LDS->global store (ASYNCcnt-tracked). Guarded: fall back to
// direct b128 stores if this toolchain doesn't declare the builtin.
#if defined(__HIP_DEVICE_COMPILE__) && defined(__has_builtin)
#if __has_builtin(__builtin_amdgcn_global_store_async_from_lds_b128)
#define SH_HAVE_ASYNC 1
#endif
#endif
#ifndef SH_HAVE_ASYNC
#define SH_HAVE_ASYNC 0
#endif

// Exact pointer types the async builtin expects (from hipcc's diagnostic):
// a pointer to a 16-byte int vector, in AS1 (global) / AS3 (LDS).
typedef int sh_v4i __attribute__((vector_size(4 * sizeof(int))));
typedef __attribute__((address_space(1))) sh_v4i* sh_gptr;
typedef __attribute__((address_space(3))) sh_v4i* sh_lptr;

__global__ __launch_bounds__(256) void sh16_encode_kernel(
    const float* __restrict__ in,   // [B,3] fp32
    const int*   __restrict__ szp,  // scalar divisor
    float*       __restrict__ out,  // [B,16] fp32
    int B)
{
    const int t = (int)threadIdx.x;         // block-local point id
    const int i = (int)blockIdx.x * 256 + t;

    const float inv = 1.0f / (float)szp[0]; // uniform scalar (s_load + v_rcp once)

    float x = 0.0f, y = 0.0f, z = 0.0f;
    if (i < B) {
        const float* p = in + (size_t)3 * (size_t)i;
        x = p[0] * inv;
        y = p[1] * inv;
        z = p[2] * inv;
    }
    const float xx = x * x, yy = y * y, zz = z * z;
    const float xy = x * y, yz = y * z, xz = x * z;

    // 16 outputs as four float4 chunks, in reference order
    const float4 c0 = make_float4(SH_C0,
                                  -SH_C1 * y,
                                  SH_C1 * z,
                                  -SH_C1 * x);
    const float4 c1 = make_float4(SH_C2_0 * xy,
                                  -SH_C2_0 * yz,
                                  SH_C2_2a * zz - SH_C2_2b,
                                  -SH_C2_0 * xz);
    const float4 c2 = make_float4(SH_C2_4 * (xx - yy),
                                  SH_C3_0 * y * (-3.0f * xx + yy),
                                  SH_C3_1 * xy * z,
                                  SH_C3_2 * y * (1.0f - 5.0f * zz));
    const float4 c3 = make_float4(SH_C3_3 * z * (5.0f * zz - 3.0f),
                                  SH_C3_2 * x * (1.0f - 5.0f * zz),
                                  SH_C3_5 * z * (xx - yy),
                                  SH_C3_0 * x * (-xx + 3.0f * yy));

#if SH_HAVE_ASYNC
    const bool full_block = ((int)blockIdx.x * 256 + 256) <= B; // block-uniform
    if (full_block) {
        // 16 KB staging: slot (point, chunk) with chunk slot rotated by
        // (t>>2)&3 so ds_store_b128 bank conflicts drop from 8-way to 2-way.
        __shared__ float4 lds4[256 * 4];
        const int rot = (t >> 2) & 3;
        float4* lbase = &lds4[t * 4];
        lbase[(0 + rot) & 3] = c0;
        lbase[(1 + rot) & 3] = c1;
        lbase[(2 + rot) & 3] = c2;
        lbase[(3 + rot) & 3] = c3;

        // Each wave drains only its own LDS region -> wave-local DS wait,
        // no workgroup barrier required.
        asm volatile("s_wait_dscnt 0x0" ::: "memory");

        const int w = t >> 5;   // wave in block (8 waves of 32)
        const int L = t & 31;   // lane
        float4* gbase = (float4*)(out + (size_t)blockIdx.x * (256 * 16));
        #pragma unroll
        for (int k = 0; k < 4; ++k) {
            const int q    = k * 32 + L;            // float4 idx in wave's 2KB region
            const int p    = w * 32 + (q >> 2);     // source point (block-local)
            const int slot = ((q & 3) + ((p >> 2) & 3)) & 3; // undo swizzle
            float4* src = &lds4[p * 4 + slot];
            float4* dst = gbase + (w * 128 + q);    // 512B contiguous per wave/instr
            __builtin_amdgcn_global_store_async_from_lds_b128(
                (sh_gptr)dst, (sh_lptr)src, 0, 0);
        }
        // S_ENDPGM performs implicit wait-idle -> ASYNCcnt drains before exit.
        return;
    }
#endif
    // Fallback / tail-block path: direct 4x b128 stores.
    if (i < B) {
        float4* o = (float4*)(out + (size_t)i * 16);
        o[0] = c0; o[1] = c1; o[2] = c2; o[3] = c3;
    }
}

extern "C" void kernel_launch(void* const* d_in, const int* in_sizes, int n_in,
                              void* d_out, int out_size, void* d_ws, size_t ws_size,
                              hipStream_t stream) {
    const float* in  = (const float*)d_in[0];
    const int*   szp = (const int*)d_in[1];
    float*       out = (float*)d_out;

    const int B = in_sizes[0] / 3;          // points
    const int blocks = (B + 255) / 256;
    if (blocks > 0) {
        sh16_encode_kernel<<<blocks, 256, 0, stream>>>(in, szp, out, B);
    }
}